// VoxelGraph_128849019530
// MI455X (gfx1250) — compile-verified
//
#include <hip/hip_runtime.h>
#include <hip/hip_bf16.h>
#include <float.h>

// Problem constants (match reference)
#define N_REF   16384
#define N_QUERY 8192
#define KNN     32
#define QT      16              // queries per block (one WMMA M-tile)
#define WAVES   4               // waves per block
#define CH      128             // refs per selection chunk
#define CHP     132             // padded LDS stride: 132%64==4 -> conflict-free,
                                // 132*4B=528B keeps float4 rows 16B-aligned
#define NCHUNK  (N_REF / CH)    // 128
#define TILES   (CH / 16)       // 8

typedef __attribute__((ext_vector_type(2))) float v2f;
typedef __attribute__((ext_vector_type(8))) float v8f;

// ---------------------------------------------------------------------------
// Kernel 0: pmin[0..2] = min over all points of (x,y,z) - 2.0  (vs = 1.0)
// ---------------------------------------------------------------------------
__global__ __launch_bounds__(256) void vg_minred(const float* __restrict__ ref,
                                                 const float* __restrict__ qry,
                                                 float* __restrict__ pmin)
{
    __shared__ float sx[256], sy[256], sz[256];
    const int tid = threadIdx.x;
    float mx = FLT_MAX, my = FLT_MAX, mz = FLT_MAX;
    const float4* rv = (const float4*)ref;
    const float4* qv = (const float4*)qry;
    for (int i = tid; i < N_REF; i += 256) {
        float4 p = rv[i];
        mx = fminf(mx, p.y); my = fminf(my, p.z); mz = fminf(mz, p.w);
    }
    for (int i = tid; i < N_QUERY; i += 256) {
        float4 p = qv[i];
        mx = fminf(mx, p.y); my = fminf(my, p.z); mz = fminf(mz, p.w);
    }
    sx[tid] = mx; sy[tid] = my; sz[tid] = mz;
    __syncthreads();
    for (int off = 128; off > 0; off >>= 1) {
        if (tid < off) {
            sx[tid] = fminf(sx[tid], sx[tid + off]);
            sy[tid] = fminf(sy[tid], sy[tid + off]);
            sz[tid] = fminf(sz[tid], sz[tid + off]);
        }
        __syncthreads();
    }
    if (tid == 0) {
        pmin[0] = sx[0] - 2.0f;
        pmin[1] = sy[0] - 2.0f;
        pmin[2] = sz[0] - 2.0f;
    }
}

// ---------------------------------------------------------------------------
// Kernel 1: WMMA distance tiles + masked exact top-32 per query.
//   d' = |r|^2 - 2 q.r   (d2 minus per-query constant |q|^2 -> same ordering)
//   A(16x4) row m = [qx qy qz 1]; B(4x16) col n = [-2rx -2ry -2rz |r|^2]
//   A frag (lane L, m = L%16): v[0] = K(0|2), v[1] = K(1|3)  (halves K0,K1 | K2,K3)
//   C frag: VGPR i -> M = i + (L>=16 ? 8 : 0), N = L%16
// ---------------------------------------------------------------------------
__global__ __launch_bounds__(128) void vg_knn(const float* __restrict__ ref,
                                              const float* __restrict__ qry,
                                              const float* __restrict__ pmin,
                                              int* __restrict__ out_ref,
                                              int* __restrict__ out_q)
{
    __shared__ float qraw[QT][4];                             // b,x,y,z per query
    __shared__ int   qvox[QT][4];                             // b, vx, vy, vz
    __shared__ __align__(16) float chunkD[WAVES][QT][CHP];    // staged masked dists
    __shared__ float topD[WAVES][QT][KNN];
    __shared__ int   topI[WAVES][QT][KNN];

    const int  tid   = threadIdx.x;
    const int  wave  = tid >> 5;
    const int  lane  = tid & 31;
    const int  m     = lane & 15;
    const bool hi    = lane >= 16;
    const int  qbase = blockIdx.x * QT;
    const float INF  = __builtin_inff();

    const float px = pmin[0], py = pmin[1], pz = pmin[2];

    if (tid < QT) {
        float4 q = ((const float4*)qry)[qbase + tid];
        qraw[tid][0] = q.x; qraw[tid][1] = q.y; qraw[tid][2] = q.z; qraw[tid][3] = q.w;
        qvox[tid][0] = __float2int_rn(q.x);        // batch: equality only
        qvox[tid][1] = __float2int_rn(q.y - px);   // voxel coords (+1 offset cancels)
        qvox[tid][2] = __float2int_rn(q.z - py);
        qvox[tid][3] = __float2int_rn(q.w - pz);
    }
    if (lane < QT) {
#pragma unroll
        for (int k = 0; k < KNN; ++k) {
            topD[wave][lane][k] = INF;
            topI[wave][lane][k] = -1;
        }
    }
    __syncthreads();

    // Hoisted A fragment for this lane's query column
    v2f a;
    a[0] = hi ? qraw[m][3] : qraw[m][1];   // K0 = qx  | K2 = qz
    a[1] = hi ? 1.0f       : qraw[m][2];   // K1 = qy  | K3 = 1
    // Hoisted voxel data for this lane's 8 query rows of the C fragment
    int qb8[8], qx8[8], qy8[8], qz8[8];
#pragma unroll
    for (int i = 0; i < 8; ++i) {
        const int r = i + (hi ? 8 : 0);
        qb8[i] = qvox[r][0]; qx8[i] = qvox[r][1];
        qy8[i] = qvox[r][2]; qz8[i] = qvox[r][3];
    }

    float curMax = INF;
    int   maxPos = 0;
    const float4* refv = (const float4*)ref;

    for (int c = wave; c < NCHUNK; c += WAVES) {
        const int base = c * CH;
#pragma unroll 2
        for (int t = 0; t < TILES; ++t) {
            const int j = base + t * 16 + m;             // this lane's ref row
            float4 rr = refv[j];
            if (t + 1 < TILES)
                __builtin_prefetch((const void*)&refv[j + 16], 0, 1);
            const float rx = rr.y, ry = rr.z, rz = rr.w;
            v2f b;
            if (hi) { b[0] = -2.0f * rz; b[1] = rx*rx + ry*ry + rz*rz; } // K2 | K3
            else    { b[0] = -2.0f * rx; b[1] = -2.0f * ry; }            // K0 | K1
            v8f cacc = {};
            cacc = __builtin_amdgcn_wmma_f32_16x16x4_f32(
                /*neg_a=*/false, a, /*neg_b=*/false, b,
                /*c_mod=*/(short)0, cacc, /*reuse_a=*/false, /*reuse_b=*/false);

            const int rb  = __float2int_rn(rr.x);
            const int rvx = __float2int_rn(rx - px);
            const int rvy = __float2int_rn(ry - py);
            const int rvz = __float2int_rn(rz - pz);
#pragma unroll
            for (int i = 0; i < 8; ++i) {
                int dx = rvx - qx8[i]; dx = dx < 0 ? -dx : dx;
                int dy = rvy - qy8[i]; dy = dy < 0 ? -dy : dy;
                int dz = rvz - qz8[i]; dz = dz < 0 ? -dz : dz;
                const bool ok = (rb == qb8[i]) && (dx <= 1) && (dy <= 1) && (dz <= 1);
                const float d = ok ? cacc[i] : INF;
                const int M = i + (hi ? 8 : 0);
                chunkD[wave][M][t * 16 + m] = d;
            }
        }
        __syncthreads();   // cross-lane LDS visibility (all waves iterate equally)

        if (lane < QT) {   // lane owns query `lane`: exact top-KNN via replace-max
            const float4* myChunk4 = (const float4*)&chunkD[wave][lane][0];
            float* myD = &topD[wave][lane][0];
            int*   myI = &topI[wave][lane][0];
            for (int j4 = 0; j4 < CH / 4; ++j4) {
                const float4 v4 = myChunk4[j4];               // ds_load_b128
                const float g = fminf(fminf(v4.x, v4.y), fminf(v4.z, v4.w));
                if (g < curMax) {                              // rare slow path
                    const int jb = base + j4 * 4;
                    const float vv[4] = { v4.x, v4.y, v4.z, v4.w };
#pragma unroll
                    for (int u = 0; u < 4; ++u) {
                        const float d = vv[u];
                        if (d < curMax) {    // strict < keeps lower ref idx on ties
                            myD[maxPos] = d;
                            myI[maxPos] = jb + u;
                            float mx = myD[0]; int mp = 0;
#pragma unroll
                            for (int k = 1; k < KNN; ++k) {
                                const float v = myD[k];
                                if (v > mx) { mx = v; mp = k; }
                            }
                            curMax = mx; maxPos = mp;
                        }
                    }
                }
            }
        }
        __syncthreads();
    }

    // Sort each wave's lists ascending by (d, idx) — matches top_k tie order
    if (lane < QT) {
        float* myD = &topD[wave][lane][0];
        int*   myI = &topI[wave][lane][0];
        for (int k = 0; k < KNN - 1; ++k) {
            float bd = myD[k]; int bi = myI[k]; int bp = k;
            for (int j2 = k + 1; j2 < KNN; ++j2) {
                const float d = myD[j2]; const int ix = myI[j2];
                if (d < bd || (d == bd && ix < bi)) { bd = d; bi = ix; bp = j2; }
            }
            if (bp != k) {
                myD[bp] = myD[k]; myI[bp] = myI[k];
                myD[k] = bd; myI[k] = bi;
            }
        }
    }
    __syncthreads();

    // Final exact 4-way merge of sorted per-wave lists (wave 0, lanes 0..15)
    if (tid < QT) {
        int p0 = 0, p1 = 0, p2 = 0, p3 = 0;
        const int q = qbase + tid;
        for (int k = 0; k < KNN; ++k) {
            float d0 = (p0 < KNN) ? topD[0][tid][p0] : INF;
            int   i0 = (p0 < KNN) ? topI[0][tid][p0] : 0x7fffffff;
            float d1 = (p1 < KNN) ? topD[1][tid][p1] : INF;
            int   i1 = (p1 < KNN) ? topI[1][tid][p1] : 0x7fffffff;
            float d2 = (p2 < KNN) ? topD[2][tid][p2] : INF;
            int   i2 = (p2 < KNN) ? topI[2][tid][p2] : 0x7fffffff;
            float d3 = (p3 < KNN) ? topD[3][tid][p3] : INF;
            int   i3 = (p3 < KNN) ? topI[3][tid][p3] : 0x7fffffff;
            int src = 0; float bd = d0; int bi = i0;
            if (d1 < bd || (d1 == bd && i1 < bi)) { src = 1; bd = d1; bi = i1; }
            if (d2 < bd || (d2 == bd && i2 < bi)) { src = 2; bd = d2; bi = i2; }
            if (d3 < bd || (d3 == bd && i3 < bi)) { src = 3; bd = d3; bi = i3; }
            out_ref[q * KNN + k] = (bd == INF) ? -1 : bi;
            out_q[q * KNN + k]   = q;
            if      (src == 0) ++p0;
            else if (src == 1) ++p1;
            else if (src == 2) ++p2;
            else               ++p3;
        }
    }
}

// ---------------------------------------------------------------------------
extern "C" void kernel_launch(void* const* d_in, const int* in_sizes, int n_in,
                              void* d_out, int out_size, void* d_ws, size_t ws_size,
                              hipStream_t stream) {
    (void)in_sizes; (void)n_in; (void)out_size; (void)ws_size;
    const float* ref = (const float*)d_in[0];   // ref_bxyz  (N_REF, 4) f32
    const float* qry = (const float*)d_in[1];   // query_bxyz(N_QUERY,4) f32
    float* pmin = (float*)d_ws;                 // 3 floats of scratch
    int* outp = (int*)d_out;                    // edges (2, N_QUERY*KNN) int32

    vg_minred<<<1, 256, 0, stream>>>(ref, qry, pmin);
    vg_knn<<<N_QUERY / QT, 128, 0, stream>>>(ref, qry, pmin,
                                             outp, outp + N_QUERY * KNN);
}